// S_CEMBlock_wjq_77567109365965
// MI455X (gfx1250) — compile-verified
//
#include <hip/hip_runtime.h>
#include <hip/hip_bf16.h>

#define B_   2
#define C_   48
#define HW_  4096
#define C3_  144
#define C2_  96

typedef __attribute__((ext_vector_type(2))) float v2f;
typedef __attribute__((ext_vector_type(8))) float v8f;

static __device__ __forceinline__ v8f wmma4(v2f a, v2f b, v8f c) {
    // D(16x16,f32) = A(16x4,f32) x B(4x16,f32) + C   -- exact fp32 matrix op
    return __builtin_amdgcn_wmma_f32_16x16x4_f32(false, a, false, b, (short)0, c, false, false);
}

// async copy 16B global -> LDS, tracked by ASYNCcnt (CDNA5)
static __device__ __forceinline__ void async_cp16(unsigned lds_byte, const float* g) {
    asm volatile("global_load_async_to_lds_b128 %0, %1, off"
                 :: "v"(lds_byte), "v"((unsigned long long)(size_t)g) : "memory");
}
static __device__ __forceinline__ void wait_async1() {
    asm volatile("s_wait_asynccnt 0x1" ::: "memory");
}

// ---------------- per-pixel LayerNorm over C (biased var), channels-first ----
__global__ void ln_k(const float* __restrict__ x, const float* __restrict__ w,
                     const float* __restrict__ bb, float* __restrict__ y) {
    int idx = blockIdx.x * blockDim.x + threadIdx.x;   // < B_*HW_
    int b = idx >> 12, p = idx & 4095;
    const float* xb = x + (size_t)b * C_ * HW_ + p;
    float vals[C_]; float s = 0.f, ss = 0.f;
#pragma unroll
    for (int c = 0; c < C_; ++c) { float v = xb[c * HW_]; vals[c] = v; s += v; ss += v * v; }
    float mu = s * (1.f / C_);
    float var = ss * (1.f / C_) - mu * mu;
    float r = rsqrtf(var + 1e-6f);
    float* yb = y + (size_t)b * C_ * HW_ + p;
#pragma unroll
    for (int c = 0; c < C_; ++c) yb[c * HW_] = w[c] * ((vals[c] - mu) * r) + bb[c];
}

// ---------------- global average pool per (b,c) ------------------------------
__global__ void pool_k(const float* __restrict__ xn, float* __restrict__ pooled) {
    __shared__ float red[256];
    int bc = blockIdx.x;                               // < B_*C_
    const float* p = xn + (size_t)bc * HW_;
    float s = 0.f;
    for (int i = threadIdx.x; i < HW_; i += 256) s += p[i];
    red[threadIdx.x] = s; __syncthreads();
    for (int st = 128; st > 0; st >>= 1) {
        if (threadIdx.x < st) red[threadIdx.x] += red[threadIdx.x + st];
        __syncthreads();
    }
    if (threadIdx.x == 0) pooled[bc] = red[0] * (1.f / HW_);
}

// ---------------- SE squeeze-excite (C->3->C) --------------------------------
__global__ void se_k(const float* __restrict__ pooled,
                     const float* __restrict__ w1, const float* __restrict__ b1,
                     const float* __restrict__ w2, const float* __restrict__ b2,
                     float* __restrict__ scale) {
    int b = blockIdx.x;
    __shared__ float ys[3];
    const float* pb = pooled + b * C_;
    if (threadIdx.x < 3) {
        float s = b1[threadIdx.x];
        for (int c = 0; c < C_; ++c) s += w1[threadIdx.x * C_ + c] * pb[c];
        ys[threadIdx.x] = fmaxf(s, 0.f);
    }
    __syncthreads();
    if (threadIdx.x < C_) {
        float s = b2[threadIdx.x];
        for (int j = 0; j < 3; ++j) s += w2[threadIdx.x * 3 + j] * ys[j];
        scale[b * C_ + threadIdx.x] = 1.f / (1.f + __expf(-s));
    }
}

// ---------------- 1x1 conv as WMMA f32 GEMM ----------------------------------
template <int MODE>
__global__ void gemm_k(const float* __restrict__ Wt, const float* __restrict__ bias,
                       const float* __restrict__ X, float* __restrict__ Y, int M,
                       const float* __restrict__ resid, const float* __restrict__ gamma) {
    int wave = threadIdx.x >> 5, lane = threadIdx.x & 31;
    int hi = lane >> 4, ln = lane & 15;
    int Mt = M >> 4;
    int tile = blockIdx.x * (blockDim.x >> 5) + wave;
    int perb = Mt * 256;
    int b = tile / perb; int rem = tile - b * perb;
    int m0 = (rem >> 8) << 4, p0 = (rem & 255) << 4;
    const float* Xb = X + (size_t)b * (MODE == 0 ? C_ : 2 * C_) * HW_;
    int p = p0 + ln;
    v8f acc;
#pragma unroll
    for (int r = 0; r < 8; ++r) acc[r] = bias[m0 + r + 8 * hi];
#pragma unroll
    for (int kk = 0; kk < 12; ++kk) {                    // K = 48 = 12 x 4
        int ka = kk * 4 + 2 * hi;
        v2f a, bf;
        a.x = Wt[(m0 + ln) * C_ + ka];
        a.y = Wt[(m0 + ln) * C_ + ka + 1];
        if (MODE == 0) {
            bf.x = Xb[ka * HW_ + p];
            bf.y = Xb[(ka + 1) * HW_ + p];
        } else {
            bf.x = Xb[ka * HW_ + p] * Xb[(ka + C_) * HW_ + p];
            bf.y = Xb[(ka + 1) * HW_ + p] * Xb[(ka + 1 + C_) * HW_ + p];
        }
        acc = wmma4(a, bf, acc);
    }
    if (MODE == 0) {
        float* Yb = Y + (size_t)b * M * HW_;
#pragma unroll
        for (int r = 0; r < 8; ++r) Yb[(m0 + r + 8 * hi) * HW_ + p] = acc[r];
    } else {
        const float* rb = resid + (size_t)b * C_ * HW_;
        float* Yb = Y + (size_t)b * C_ * HW_;
#pragma unroll
        for (int r = 0; r < 8; ++r) {
            int row = m0 + r + 8 * hi;
            Yb[row * HW_ + p] = rb[row * HW_ + p] + acc[r] * gamma[row];
        }
    }
}

// ---------------- depthwise 3x3, SAME ---------------------------------------
__global__ void dw_k(const float* __restrict__ X, const float* __restrict__ w,
                     const float* __restrict__ bb, float* __restrict__ Y) {
    int idx = blockIdx.x * 256 + threadIdx.x;            // < B_*C3_*HW_
    int p = idx & 4095; int oc = (idx >> 12) % C3_; int b = idx / (C3_ * HW_);
    int h = p >> 6, wv = p & 63;
    const float* xb = X + (size_t)(b * C3_ + oc) * HW_;
    const float* wk = w + oc * 9;
    float s = bb[oc];
#pragma unroll
    for (int di = -1; di <= 1; ++di) {
        int hh = h + di; if (hh < 0 || hh > 63) continue;
#pragma unroll
        for (int dj = -1; dj <= 1; ++dj) {
            int ww = wv + dj; if (ww < 0 || ww > 63) continue;
            s += wk[(di + 1) * 3 + (dj + 1)] * xb[hh * 64 + ww];
        }
    }
    Y[idx] = s;
}

// ---------------- pack: l2-normalize Q (fold temp) and K, token-major --------
__global__ void pack_k(const float* __restrict__ qkv, const float* __restrict__ temp,
                       float* __restrict__ qp, float* __restrict__ kp) {
    int idx = blockIdx.x * 256 + threadIdx.x;            // < B_*3*HW_
    int bh = idx >> 12, token = idx & 4095;
    int b = bh / 3, head = bh - b * 3;
    const float* qc = qkv + (size_t)b * C3_ * HW_ + (head * 16) * HW_ + token;
    const float* kc = qc + C_ * HW_;
    float q[16], k[16]; float qs = 0.f, ks = 0.f;
#pragma unroll
    for (int d = 0; d < 16; ++d) {
        float a = qc[d * HW_]; q[d] = a; qs += a * a;
        float c = kc[d * HW_]; k[d] = c; ks += c * c;
    }
    float qi = temp[head] * rsqrtf(fmaxf(qs, 1e-24f));
    float ki = rsqrtf(fmaxf(ks, 1e-24f));
    float* qo = qp + (size_t)idx * 16;
    float* ko = kp + (size_t)idx * 16;
#pragma unroll
    for (int d = 0; d < 16; ++d) { qo[d] = q[d] * qi; ko[d] = k[d] * ki; }
}

// ---------------- flash attention + fused residual mix -----------------------
// Scores bounded (|q_hat.k_hat|<=1) => exp needs no max subtraction.
// S^T = K_hat x Q_hat^T; O^T = V^T x P^T; l = ones x P^T (via WMMA).
// K_hat/V tiles staged to LDS with async b128 copies, TRIPLE-buffered,
// single barrier per iteration (wait -> barrier -> issue -> compute).
// Epilogue writes y1 = x + xn*se*beta + O*beta2 directly.
__global__ void attn_k(const float* __restrict__ qkv, const float* __restrict__ kp,
                       const float* __restrict__ qp,
                       const float* __restrict__ x, const float* __restrict__ xn,
                       const float* __restrict__ sescale, const float* __restrict__ beta,
                       const float* __restrict__ beta2, float* __restrict__ y1) {
    __shared__ float smem[1536];                         // kbuf[3][256] | vbuf[3][256]
    int tid = threadIdx.x;
    int wave = tid >> 5, lane = tid & 31;
    int hi = lane >> 4, ln = lane & 15;
    int gw = blockIdx.x * 8 + wave;                      // global row-tile id
    int bh = gw >> 8;                                    // b*3+head, uniform in block
    int m0 = (gw & 255) << 4;
    int b = bh / 3, head = bh - b * 3;
    const float* Vc = qkv + (size_t)b * C3_ * HW_ + (2 * C_ + head * 16) * HW_;
    const float* kpb = kp + (size_t)bh * HW_ * 16;
    unsigned sb = (unsigned)(unsigned long long)(void*)smem;

    // Q row (query m0+ln), prenormalized with temp folded: contiguous 64B
    const float* qr = qp + ((size_t)bh * HW_ + m0 + ln) * 16;
    float qrow[16];
#pragma unroll
    for (int d = 0; d < 16; ++d) qrow[d] = qr[d];
    v2f qb[4];
#pragma unroll
    for (int j = 0; j < 4; ++j) { qb[j].x = qrow[4 * j + 2 * hi]; qb[j].y = qrow[4 * j + 2 * hi + 1]; }

    v2f ones; ones.x = 1.f; ones.y = 1.f;
    v8f acc, lacc;
#pragma unroll
    for (int r = 0; r < 8; ++r) { acc[r] = 0.f; lacc[r] = 0.f; }

    // issuer threads: tid<64 stage K tile (1KB, [token][d]); tid in [64,128) stage V ([d][token])
    int ci = tid & 63;
    int vd = ci >> 2, vt4 = (ci & 3) * 4;
    // prologue: stage tiles 0,1 into slots 0,1
    if (tid < 64) {
        async_cp16(sb + ci * 16, kpb + ci * 4);
        async_cp16(sb + 1024 + ci * 16, kpb + (size_t)(16 * 16) + ci * 4);
    } else if (tid < 128) {
        async_cp16(sb + 3072 + ci * 16, Vc + (size_t)vd * HW_ + vt4);
        async_cp16(sb + 3072 + 1024 + ci * 16, Vc + (size_t)vd * HW_ + 16 + vt4);
    }

    int sCur = 0, sIss = 2;
    for (int nt = 0; nt < 256; ++nt) {
        wait_async1();                                   // tile nt landed (in-order)
        __syncthreads();                                 // visible to all; fences slot reuse
        int nn = nt + 2; nn = (nn >= 256) ? nn - 256 : nn;   // uniform wrap
        if (tid < 64)
            async_cp16(sb + sIss * 1024 + ci * 16, kpb + (size_t)(nn << 4) * 16 + ci * 4);
        else if (tid < 128)
            async_cp16(sb + 3072 + sIss * 1024 + ci * 16, Vc + (size_t)vd * HW_ + (nn << 4) + vt4);
        const float* kb = smem + sCur * 256;             // [key][d]
        const float* vb = smem + 768 + sCur * 256;       // [d][token]
        v8f S;
#pragma unroll
        for (int r = 0; r < 8; ++r) S[r] = 0.f;
#pragma unroll
        for (int j = 0; j < 4; ++j) {
            v2f ka; ka.x = kb[ln * 16 + 4 * j + 2 * hi]; ka.y = kb[ln * 16 + 4 * j + 2 * hi + 1];
            S = wmma4(ka, qb[j], S);                     // S^T tile = K_hat Q_hat^T
        }
        // P = exp(relu(S)), bounded -> no max tracking
        float P[8], T[8];
#pragma unroll
        for (int r = 0; r < 8; ++r) P[r] = __expf(fmaxf(S[r], 0.f));
#pragma unroll
        for (int r = 0; r < 8; ++r) T[r] = __shfl_xor(P[r], 16, 32);   // half-swap
        v2f pf[4];
        pf[0].x = hi ? T[2] : P[0]; pf[0].y = hi ? T[3] : P[1];
        pf[1].x = hi ? T[6] : P[4]; pf[1].y = hi ? T[7] : P[5];
        pf[2].x = hi ? P[2] : T[0]; pf[2].y = hi ? P[3] : T[1];
        pf[3].x = hi ? P[6] : T[4]; pf[3].y = hi ? P[7] : T[5];
#pragma unroll
        for (int j = 0; j < 4; ++j) {
            v2f va; va.x = vb[ln * 16 + 4 * j + 2 * hi]; va.y = vb[ln * 16 + 4 * j + 2 * hi + 1];
            acc = wmma4(va, pf[j], acc);                 // O^T += V^T P^T
        }
#pragma unroll
        for (int j = 0; j < 4; ++j) lacc = wmma4(ones, pf[j], lacc);   // l += 1 . P^T
        sCur = (sCur == 2) ? 0 : sCur + 1;
        sIss = (sIss == 2) ? 0 : sIss + 1;
    }
    // epilogue: fused  y1 = x + xn*se*beta + (O/l)*beta2
    float linv = 1.0f / lacc[0];
    int pix = m0 + ln;
    int c0 = head * 16 + 8 * hi;
    const float* xB  = x  + (size_t)b * C_ * HW_;
    const float* xnB = xn + (size_t)b * C_ * HW_;
    float* yB = y1 + (size_t)b * C_ * HW_;
#pragma unroll
    for (int r = 0; r < 8; ++r) {
        int c = c0 + r;
        size_t ix = (size_t)c * HW_ + pix;
        float o = acc[r] * linv;
        yB[ix] = xB[ix] + xnB[ix] * sescale[b * C_ + c] * beta[c] + o * beta2[c];
    }
}

extern "C" void kernel_launch(void* const* d_in, const int* in_sizes, int n_in,
                              void* d_out, int out_size, void* d_ws, size_t ws_size,
                              hipStream_t stream) {
    const float* x      = (const float*)d_in[0];
    const float* ln1_w  = (const float*)d_in[1];
    const float* ln1_b  = (const float*)d_in[2];
    const float* qkv_w  = (const float*)d_in[3];
    const float* qkv_b  = (const float*)d_in[4];
    const float* dw_w   = (const float*)d_in[5];
    const float* dw_b   = (const float*)d_in[6];
    const float* temp   = (const float*)d_in[7];
    const float* ca_w1  = (const float*)d_in[8];
    const float* ca_b1  = (const float*)d_in[9];
    const float* ca_w2  = (const float*)d_in[10];
    const float* ca_b2  = (const float*)d_in[11];
    const float* beta   = (const float*)d_in[12];
    const float* beta2  = (const float*)d_in[13];
    const float* ln2_w  = (const float*)d_in[14];
    const float* ln2_b  = (const float*)d_in[15];
    const float* c4_w   = (const float*)d_in[16];
    const float* c4_b   = (const float*)d_in[17];
    const float* c5_w   = (const float*)d_in[18];
    const float* c5_b   = (const float*)d_in[19];
    const float* gamma  = (const float*)d_in[20];
    float* out = (float*)d_out;

    float* ws = (float*)d_ws;
    const size_t nBCHW = (size_t)B_ * C_ * HW_;          // 393216
    float* xn      = ws;                                 // B*C*HW
    float* pooled  = xn + nBCHW;                         // 96 (+pad)
    float* sescale = pooled + 128;                       // 96 (+pad)
    float* qkv_pre = sescale + 128;                      // B*144*HW
    float* qkvb    = qkv_pre + (size_t)B_ * C3_ * HW_;   // B*144*HW
    float* y1      = qkvb + (size_t)B_ * C3_ * HW_;      // B*C*HW
    float* y1n     = y1 + nBCHW;                         // B*C*HW
    float* f2c     = y1n + nBCHW;                        // B*96*HW
    float* qpack   = f2c + (size_t)B_ * C2_ * HW_;       // B*3*HW*16
    float* kpack   = qpack + nBCHW;                      // B*3*HW*16

    // 1) LN1
    ln_k<<<(B_ * HW_) / 256, 256, 0, stream>>>(x, ln1_w, ln1_b, xn);
    // 2) SE pool + gate
    pool_k<<<B_ * C_, 256, 0, stream>>>(xn, pooled);
    se_k<<<B_, 64, 0, stream>>>(pooled, ca_w1, ca_b1, ca_w2, ca_b2, sescale);
    // 3) QKV 1x1 (M=144)
    gemm_k<0><<<576, 256, 0, stream>>>(qkv_w, qkv_b, xn, qkv_pre, C3_, nullptr, nullptr);
    // 4) depthwise 3x3
    dw_k<<<(B_ * C3_ * HW_) / 256, 256, 0, stream>>>(qkv_pre, dw_w, dw_b, qkvb);
    // 5) pack normalized Q/K token-major
    pack_k<<<(B_ * 3 * HW_) / 256, 256, 0, stream>>>(qkvb, temp, qpack, kpack);
    // 6) flash attention + fused residual mix (writes y1)
    attn_k<<<192, 256, 0, stream>>>(qkvb, kpack, qpack, x, xn, sescale, beta, beta2, y1);
    // 7) LN2
    ln_k<<<(B_ * HW_) / 256, 256, 0, stream>>>(y1, ln2_w, ln2_b, y1n);
    // 8) conv4 (M=96)
    gemm_k<0><<<384, 256, 0, stream>>>(c4_w, c4_b, y1n, f2c, C2_, nullptr, nullptr);
    // 9) gated conv5 + residual (M=48)
    gemm_k<1><<<192, 256, 0, stream>>>(c5_w, c5_b, f2c, out, C_, y1, gamma);
}